// Quantizer_31619549233582
// MI455X (gfx1250) — compile-verified
//
#include <hip/hip_runtime.h>
#include <stdint.h>

#define BLOCK 256
#define NCENT 16
// each lane stages 2 float4s (32B) per tile; a tile = BLOCK*2 float4s
#define V4_PER_TILE (BLOCK * 2)

// Low 32 bits of a generic pointer to a __shared__ object = wave-relative LDS
// byte offset (LDS aperture keeps the offset in addr[31:0]).
__device__ __forceinline__ uint32_t lds_off(const void* p) {
  return (uint32_t)(uintptr_t)p;
}

// nearest-center quantization of one float4 against 15 sorted midpoints
__device__ __forceinline__ void quant4(const float4 xv, const float* mid,
                                       const float* cshared, float4* hv, int4* iv) {
  const float xs[4] = {xv.x, xv.y, xv.z, xv.w};
  float h[4];
  int   id[4];
#pragma unroll
  for (int c = 0; c < 4; ++c) {
    int k = 0;
#pragma unroll
    for (int l = 0; l < NCENT - 1; ++l) k += (xs[c] > mid[l]) ? 1 : 0;
    id[c] = k;
    h[c]  = cshared[k];  // 16-word LDS gather: bank-conflict free
  }
  *hv = {h[0], h[1], h[2], h[3]};
  *iv = {id[0], id[1], id[2], id[3]};
}

__global__ __launch_bounds__(BLOCK) void quantizer_st_fwd(
    const float* __restrict__ x,
    const float* __restrict__ centers,
    float* __restrict__ out_st,
    float* __restrict__ out_hard,
    int*   __restrict__ out_idx,
    int n4,      // total float4 elements
    int nfull)   // number of full tiles handled by the async pipeline
{
  __shared__ float  cshared[NCENT];
  __shared__ float4 stage[2][V4_PER_TILE];

  const int tid = threadIdx.x;
  if (tid < NCENT) cshared[tid] = centers[tid];
  __syncthreads();

  // Midpoints between sorted centers (uniform; scalar).
  float mid[NCENT - 1];
#pragma unroll
  for (int l = 0; l < NCENT - 1; ++l)
    mid[l] = 0.5f * (centers[l] + centers[l + 1]);

  const uint32_t lds0 = lds_off(&stage[0][2 * tid]);
  const uint32_t lds1 = lds_off(&stage[1][2 * tid]);
  const unsigned long long xbase = (unsigned long long)(uintptr_t)x;
  const int G = (int)gridDim.x;

  int tile = (int)blockIdx.x;

  // Prologue: async-copy tile 0 into stage 0 (32B per lane, two b128s;
  // offset:16 applies to BOTH the global and LDS side per the TDM/async ISA).
  if (tile < nfull) {
    uint32_t voff = (uint32_t)tile * (V4_PER_TILE * 16u) + (uint32_t)tid * 32u;
    asm volatile("global_load_async_to_lds_b128 %0, %1, %2"
                 :: "v"(lds0), "v"(voff), "s"(xbase) : "memory");
    asm volatile("global_load_async_to_lds_b128 %0, %1, %2 offset:16"
                 :: "v"(lds0), "v"(voff), "s"(xbase) : "memory");
  }

  int sIdx = 0;
  for (; tile < nfull; tile += G) {
    const int next = tile + G;
    if (next < nfull) {
      uint32_t voff = (uint32_t)next * (V4_PER_TILE * 16u) + (uint32_t)tid * 32u;
      uint32_t dst  = sIdx ? lds0 : lds1;
      asm volatile("global_load_async_to_lds_b128 %0, %1, %2"
                   :: "v"(dst), "v"(voff), "s"(xbase) : "memory");
      asm volatile("global_load_async_to_lds_b128 %0, %1, %2 offset:16"
                   :: "v"(dst), "v"(voff), "s"(xbase) : "memory");
      // 4 outstanding max; <=2 left pending means current tile's pair landed
      asm volatile("s_wait_asynccnt 0x2" ::: "memory");
    } else {
      asm volatile("s_wait_asynccnt 0x0" ::: "memory");
    }

    const int q0 = tile * V4_PER_TILE + 2 * tid;  // float4 index of first elem
    const float4 xv0 = stage[sIdx][2 * tid + 0];
    const float4 xv1 = stage[sIdx][2 * tid + 1];
    float4 hv0, hv1;
    int4   iv0, iv1;
    quant4(xv0, mid, cshared, &hv0, &iv0);
    quant4(xv1, mid, cshared, &hv1, &iv1);
    reinterpret_cast<float4*>(out_st)[q0]       = hv0;  // x_st == x_hard fwd
    reinterpret_cast<float4*>(out_st)[q0 + 1]   = hv1;
    reinterpret_cast<float4*>(out_hard)[q0]     = hv0;
    reinterpret_cast<float4*>(out_hard)[q0 + 1] = hv1;
    reinterpret_cast<int4*>(out_idx)[q0]        = iv0;
    reinterpret_cast<int4*>(out_idx)[q0 + 1]    = iv1;
    sIdx ^= 1;
  }

  // Tail (empty for the harness sizes): plain vector loads, exact guard.
  const int gt = (int)blockIdx.x * BLOCK + tid;
  for (int q = nfull * V4_PER_TILE + gt; q < n4; q += G * BLOCK) {
    const float4 xv = reinterpret_cast<const float4*>(x)[q];
    float4 hv;
    int4   iv;
    quant4(xv, mid, cshared, &hv, &iv);
    reinterpret_cast<float4*>(out_st)[q]   = hv;
    reinterpret_cast<float4*>(out_hard)[q] = hv;
    reinterpret_cast<int4*>(out_idx)[q]    = iv;
  }
}

extern "C" void kernel_launch(void* const* d_in, const int* in_sizes, int n_in,
                              void* d_out, int out_size, void* d_ws, size_t ws_size,
                              hipStream_t stream) {
  const float* x       = (const float*)d_in[0];
  const float* centers = (const float*)d_in[1];
  const int N  = in_sizes[0];   // 16*64*64*64, divisible by 4
  const int n4 = N / 4;

  float* out_st   = (float*)d_out;
  float* out_hard = out_st + (size_t)N;
  int*   out_idx  = (int*)(out_st + 2 * (size_t)N);

  const int nfull = n4 / V4_PER_TILE;  // tiles fully covered by async path
  int grid = nfull < 1024 ? nfull : 1024;
  if (grid < 1) grid = 1;

  quantizer_st_fwd<<<grid, BLOCK, 0, stream>>>(x, centers, out_st, out_hard,
                                               out_idx, n4, nfull);
}